// BasicRecurrentEntityEncoder_54382875902323
// MI455X (gfx1250) — compile-verified
//
#include <hip/hip_runtime.h>

typedef __attribute__((ext_vector_type(2))) float v2f;
typedef __attribute__((ext_vector_type(8))) float v8f;

#define B_ 64
#define S_ 128
#define L_ 32
#define E_ 256
#define H_ 128   // E/2
#define D_ 128
#define K_ 20
#define STR 132  // LDS row stride (floats): 132 % 64 == 4 -> conflict-free A-frag column reads

// ---------------------------------------------------------------------------
// Kernel A: per (b,s): embedding bag-sum -> fc1 -> fc2 -> enc row.
// Also precomputes the step-invariant pieces of the recurrent cell:
//   sWall[b,s,:] = enc[b,s,:] @ W          (the broadcast term)
//   skall[b,s,k] = dot(enc[b,s,:], keys[b,k,:])   (gate's key half)
// ---------------------------------------------------------------------------
__global__ __launch_bounds__(256) void encode_kernel(
    const int* __restrict__ prg, const float* __restrict__ emb,
    const float* __restrict__ fc1w, const float* __restrict__ fc1b,
    const float* __restrict__ fc2w, const float* __restrict__ fc2b,
    const float* __restrict__ W, const float* __restrict__ keys,
    float* __restrict__ enc, float* __restrict__ sWall, float* __restrict__ skall)
{
  __shared__ float sent[E_];
  __shared__ float hid[H_];
  __shared__ float encv[D_];
  __shared__ int   idxs[L_];
  __shared__ float part[K_ * 8];

  const int bs  = blockIdx.x;       // 0..B_*S_-1
  const int b   = bs / S_;
  const int tid = threadIdx.x;

  if (tid < L_) idxs[tid] = prg[bs * L_ + tid];
  __syncthreads();

  // bag-of-words sum over L words (gathers hit L2: table is 51MB < 192MB)
  {
    float acc = 0.f;
    #pragma unroll 4
    for (int l = 0; l < L_; ++l) acc += emb[(long)idxs[l] * E_ + tid];
    sent[tid] = acc;
  }
  __syncthreads();

  if (tid < H_) {
    float acc = fc1b[tid];
    for (int e = 0; e < E_; ++e) acc = fmaf(sent[e], fc1w[e * H_ + tid], acc);
    hid[tid] = acc;
  }
  __syncthreads();

  if (tid < D_) {
    float acc = fc2b[tid];
    for (int h = 0; h < H_; ++h) acc = fmaf(hid[h], fc2w[h * D_ + tid], acc);
    encv[tid] = acc;
    enc[(long)bs * D_ + tid] = acc;
  }
  __syncthreads();

  if (tid < D_) {  // s @ W
    float acc = 0.f;
    for (int d = 0; d < D_; ++d) acc = fmaf(encv[d], W[d * D_ + tid], acc);
    sWall[(long)bs * D_ + tid] = acc;
  }
  if (tid < K_ * 8) {  // dot(s, keys[b,k,:]) partials (8 threads x 16 elems per k)
    const int k = tid >> 3, j = tid & 7;
    const float* kr = keys + ((long)b * K_ + k) * D_ + j * 16;
    float acc = 0.f;
    #pragma unroll
    for (int i = 0; i < 16; ++i) acc = fmaf(encv[j * 16 + i], kr[i], acc);
    part[tid] = acc;
  }
  __syncthreads();
  if (tid < K_) {
    float g = 0.f;
    #pragma unroll
    for (int j = 0; j < 8; ++j) g += part[tid * 8 + j];
    skall[(long)bs * K_ + tid] = g;
  }
}

// ---------------------------------------------------------------------------
// Kernel C: kV[b,k,:] = keys[b,k,:] @ V   (step-invariant, hoisted out of scan)
// ---------------------------------------------------------------------------
__global__ __launch_bounds__(256) void kv_kernel(
    const float* __restrict__ keys, const float* __restrict__ V,
    float* __restrict__ kV)
{
  __shared__ float kb[K_ * D_];
  const int b = blockIdx.x, tid = threadIdx.x;
  for (int o = tid; o < K_ * D_; o += 256) kb[o] = keys[(long)b * K_ * D_ + o];
  __syncthreads();
  for (int o = tid; o < K_ * D_; o += 256) {
    const int k = o >> 7, n = o & 127;
    float acc = 0.f;
    for (int d = 0; d < D_; ++d) acc = fmaf(kb[k * D_ + d], V[d * D_ + n], acc);
    kV[(long)b * K_ * D_ + o] = acc;
  }
}

// ---------------------------------------------------------------------------
// Kernel B: the 128-step sequential scan. One workgroup (8 wave32s) per batch.
// EVERYTHING the loop touches is preloaded into LDS (~249KB < 320KB/WGP):
// U, h, kV, plus the whole per-batch step sequence (enc, sW, sk, mask).
// The 128 dependent steps then run entirely out of LDS + the WMMA pipe with
// zero global-memory round trips on the critical path.
// ---------------------------------------------------------------------------
__global__ __launch_bounds__(256) void scan_kernel(
    const float* __restrict__ enc, const float* __restrict__ sWall,
    const float* __restrict__ skall, const float* __restrict__ kV,
    const float* __restrict__ U, const unsigned char* __restrict__ mask,
    float* __restrict__ out)
{
  extern __shared__ float smem[];
  float* U_lds   = smem;                   // D_*STR      = 16896
  float* h_lds   = U_lds   + D_ * STR;     // 32*STR      =  4224 (rows 20..31 stay zero)
  float* kV_lds  = h_lds   + 32 * STR;     // K_*STR      =  2640
  float* upd_lds = kV_lds  + K_ * STR;     // K_*STR      =  2640
  float* encS    = upd_lds + K_ * STR;     // S_*D_       = 16384
  float* swS     = encS    + S_ * D_;      // S_*D_       = 16384
  float* skS     = swS     + S_ * D_;      // S_*K_       =  2560
  float* gpart   = skS     + S_ * K_;      // K_*8        =   160
  float* gate_l  = gpart   + K_ * 8;       // 32
  float* npart   = gate_l  + 32;           // K_*8        =   160
  float* rnorm   = npart   + K_ * 8;       // 32
  int*   maskS   = (int*)(rnorm + 32);     // S_          =   128

  const int b    = blockIdx.x;
  const int tid  = threadIdx.x;
  const int lane = tid & 31;
  const int wv   = tid >> 5;        // 8 waves
  const int lm   = lane & 15;
  const int kh   = lane >> 4;       // lane half selects K pair / +8 M rows
  const int mt   = (wv * 2) >> 3;   // M tile (0 or 1)
  const int nt0  = (wv * 2) & 7;    // first of 2 N tiles for this wave
  const int mrow = mt * 16 + lm;

  // ---- one-time preload: state, weights, and the whole step sequence ----
  for (int o = tid; o < 32 * STR; o += 256) h_lds[o] = 0.f;
  for (int o = tid; o < D_ * D_; o += 256) U_lds[(o >> 7) * STR + (o & 127)] = U[o];
  for (int o = tid; o < K_ * D_; o += 256)
    kV_lds[(o >> 7) * STR + (o & 127)] = kV[(long)b * K_ * D_ + o];
  for (int o = tid; o < S_ * D_; o += 256) {
    encS[o] = enc[(long)b * S_ * D_ + o];
    swS[o]  = sWall[(long)b * S_ * D_ + o];
  }
  for (int o = tid; o < S_ * K_; o += 256) skS[o] = skall[(long)b * S_ * K_ + o];
  for (int o = tid; o < S_; o += 256) maskS[o] = mask[(long)b * S_ + o] ? 1 : 0;
  __syncthreads();

  for (int t = 0; t < S_; ++t) {
    const float* s_row  = encS + t * D_;
    const float* sw_row = swS  + t * D_;

    // gate partials: dot(s, h[k]) (key half was precomputed into skS)
    if (tid < K_ * 8) {
      const int k = tid >> 3, j = tid & 7;
      float acc = 0.f;
      #pragma unroll
      for (int i = 0; i < 16; ++i)
        acc = fmaf(s_row[j * 16 + i], h_lds[k * STR + j * 16 + i], acc);
      gpart[tid] = acc;
    }

    // h @ U : fp32 WMMA, 2 16x16 output tiles per wave, K-loop of 32 x 16x16x4
    v8f acc0 = {};
    v8f acc1 = {};
    #pragma unroll 4
    for (int kk = 0; kk < 32; ++kk) {
      const int k0 = kk * 4 + kh * 2;
      v2f a;  a[0]  = h_lds[mrow * STR + k0];
              a[1]  = h_lds[mrow * STR + k0 + 1];
      v2f b0; b0[0] = U_lds[k0 * STR + nt0 * 16 + lm];
              b0[1] = U_lds[(k0 + 1) * STR + nt0 * 16 + lm];
      v2f b1; b1[0] = U_lds[k0 * STR + (nt0 + 1) * 16 + lm];
              b1[1] = U_lds[(k0 + 1) * STR + (nt0 + 1) * 16 + lm];
      acc0 = __builtin_amdgcn_wmma_f32_16x16x4_f32(false, a, false, b0,
                                                   (short)0, acc0, false, false);
      acc1 = __builtin_amdgcn_wmma_f32_16x16x4_f32(false, a, false, b1,
                                                   (short)0, acc1, false, false);
    }
    __syncthreads();

    if (tid < K_) {
      float g = skS[t * K_ + tid];
      #pragma unroll
      for (int j = 0; j < 8; ++j) g += gpart[tid * 8 + j];
      gate_l[tid] = 1.f / (1.f + __expf(-g));
    }
    __syncthreads();

    // upd = h + gate * relu(hU + kV + sW)
    #pragma unroll
    for (int r = 0; r < 8; ++r) {
      const int M = mt * 16 + kh * 8 + r;   // C layout: vgpr r, lane-half adds 8
      if (M < K_) {
        const int N0 = nt0 * 16 + lm;
        const int N1 = N0 + 16;
        float ht0 = fmaxf(acc0[r] + kV_lds[M * STR + N0] + sw_row[N0], 0.f);
        float ht1 = fmaxf(acc1[r] + kV_lds[M * STR + N1] + sw_row[N1], 0.f);
        upd_lds[M * STR + N0] = h_lds[M * STR + N0] + gate_l[M] * ht0;
        upd_lds[M * STR + N1] = h_lds[M * STR + N1] + gate_l[M] * ht1;
      }
    }
    __syncthreads();

    // L2 norm over D per entity row
    if (tid < K_ * 8) {
      const int k = tid >> 3, j = tid & 7;
      float acc = 0.f;
      #pragma unroll
      for (int i = 0; i < 16; ++i) {
        const float u = upd_lds[k * STR + j * 16 + i];
        acc = fmaf(u, u, acc);
      }
      npart[tid] = acc;
    }
    __syncthreads();
    if (tid < K_) {
      float ss = 0.f;
      #pragma unroll
      for (int j = 0; j < 8; ++j) ss += npart[tid * 8 + j];
      rnorm[tid] = rsqrtf(fmaxf(ss, 1e-12f));
    }
    __syncthreads();

    if (maskS[t]) {  // masked update: h = normalize(upd) only where sent_mask
      for (int o = tid; o < K_ * D_; o += 256) {
        const int k = o >> 7, n = o & 127;
        h_lds[k * STR + n] = upd_lds[k * STR + n] * rnorm[k];
      }
    }
    __syncthreads();
  }

  for (int o = tid; o < K_ * D_; o += 256)
    out[(long)b * K_ * D_ + o] = h_lds[(o >> 7) * STR + (o & 127)];
}

extern "C" void kernel_launch(void* const* d_in, const int* in_sizes, int n_in,
                              void* d_out, int out_size, void* d_ws, size_t ws_size,
                              hipStream_t stream) {
  const int*           prg  = (const int*)d_in[0];            // prgrph
  const unsigned char* mask = (const unsigned char*)d_in[1];  // sent_mask (bool)
  const float* keys = (const float*)d_in[2];
  const float* emb  = (const float*)d_in[3];
  const float* fc1w = (const float*)d_in[4];
  const float* fc1b = (const float*)d_in[5];
  const float* fc2w = (const float*)d_in[6];
  const float* fc2b = (const float*)d_in[7];
  const float* U    = (const float*)d_in[8];
  const float* V    = (const float*)d_in[9];
  const float* W    = (const float*)d_in[10];

  float* ws    = (float*)d_ws;
  float* enc   = ws;                       // B*S*D
  float* sWall = enc   + (long)B_ * S_ * D_;
  float* skall = sWall + (long)B_ * S_ * D_;
  float* kV    = skall + (long)B_ * S_ * K_;

  encode_kernel<<<B_ * S_, 256, 0, stream>>>(prg, emb, fc1w, fc1b, fc2w, fc2b,
                                             W, keys, enc, sWall, skall);
  kv_kernel<<<B_, 256, 0, stream>>>(keys, V, kV);

  const size_t smem = (size_t)(D_ * STR + 32 * STR + 2 * K_ * STR +   // U,h,kV,upd
                               2 * S_ * D_ + S_ * K_ +                // encS,swS,skS
                               2 * (K_ * 8) + 64 + S_)                // reductions+mask
                      * sizeof(float);
  scan_kernel<<<B_, 256, smem, stream>>>(enc, sWall, skall, kV, U, mask,
                                         (float*)d_out);
}